// FocalLoss_ArcFace_50148038148557
// MI455X (gfx1250) — compile-verified
//
#include <hip/hip_runtime.h>
#include <math.h>

// FocalLoss + ArcFace margin, fused.
//   inputs:  (B=4096, C=32768) fp32,  targets: (B,) int
//   output:  scalar fp32 mean loss
//
// MI455X strategy: pure HBM-bandwidth problem (512 MB read once ~= 22us at
// 23.3 TB/s). One workgroup per row; the 128 KB row is DMA'd HBM->LDS exactly
// once via global_load_async_to_lds_b128 (ASYNCcnt), with staged
// s_wait_asynccnt waits so the sum-of-squares pass overlaps the transfer
// (async loads complete in order, and each thread consumes only the bytes it
// issued). Both row reductions then run out of CDNA5's 320KB-per-WGP LDS.

#define SCALE_F 30.0f
#define COS_M   0.95533648912560601964f   // cos(0.3)
#define SIN_M   0.29552020666133957511f   // sin(0.3)

__device__ __forceinline__ float wave_reduce_sum(float v) {
#pragma unroll
    for (int o = 16; o > 0; o >>= 1)
        v += __shfl_xor(v, o, 32);
    return v;
}

// Deterministic block-wide sum. scratch must hold >= 64 floats.
__device__ __forceinline__ float block_reduce_sum(float v, float* scratch) {
    const int tid  = threadIdx.x;
    const int lane = tid & 31;
    const int wid  = tid >> 5;
    const int nw   = (int)(blockDim.x >> 5);
    v = wave_reduce_sum(v);
    if (lane == 0) scratch[wid] = v;
    __syncthreads();
    if (wid == 0) {
        float x = (lane < nw) ? scratch[lane] : 0.0f;
        x = wave_reduce_sum(x);
        if (lane == 0) scratch[63] = x;
    }
    __syncthreads();
    const float r = scratch[63];
    __syncthreads();
    return r;
}

__device__ __forceinline__ void row_tail(float sumexp, float xt, float s,
                                         float norm, float* rowloss, int row) {
    // margin-adjusted target logit: 30 * cos(acos(clip(c)) + 0.3)
    float c = xt / norm;
    c = fminf(fmaxf(c, -1.0f + 1e-7f), 1.0f - 1e-7f);
    const float m = SCALE_F * (c * COS_M - sqrtf(fmaxf(1.0f - c * c, 0.0f)) * SIN_M);
    sumexp += expf(m) - expf(s * xt);        // swap target term for margin term
    const float ce  = logf(sumexp) - m;      // -log_softmax at target
    const float pt  = expf(-ce);
    const float omp = 1.0f - pt;
    rowloss[row] = omp * omp * ce;           // gamma = 2
}

// ---------------- specialized kernel: C = 32768, 1024 threads ----------------
__global__ void __launch_bounds__(1024)
arcface_focal_row_32k(const float* __restrict__ in,
                      const int*  __restrict__ tgt,
                      float* __restrict__ rowloss)
{
    constexpr int C = 32768;
    extern __shared__ float smem[];          // [C] row cache + [64] scratch
    float* scratch = smem + C;

    const int row = blockIdx.x;
    const int tid = threadIdx.x;
    const float* rowp = in + (size_t)row * (size_t)C;

    // Base addresses: thread's first 16B chunk. Instruction IOFFSET advances
    // BOTH the LDS and global addresses (ISA 08 4.4), so 8 DMA ops share one
    // LDS-addr VGPR + one global-addr VGPR pair. Block stride = 1024*16B.
    unsigned lds0 = (unsigned)(unsigned long long)(const void*)(smem + 4 * tid);
    unsigned long long g0 = (unsigned long long)(const void*)(rowp + 4 * tid);

    asm volatile(
        "global_load_async_to_lds_b128 %0, %1, off\n\t"
        "global_load_async_to_lds_b128 %0, %1, off offset:16384\n\t"
        "global_load_async_to_lds_b128 %0, %1, off offset:32768\n\t"
        "global_load_async_to_lds_b128 %0, %1, off offset:49152\n\t"
        "global_load_async_to_lds_b128 %0, %1, off offset:65536\n\t"
        "global_load_async_to_lds_b128 %0, %1, off offset:81920\n\t"
        "global_load_async_to_lds_b128 %0, %1, off offset:98304\n\t"
        "global_load_async_to_lds_b128 %0, %1, off offset:114688"
        :: "v"(lds0), "v"(g0) : "memory");

    // ---- pass 1: sum of squares, overlapped with the DMA tail.
    // Async loads complete in order per wave; each thread reads only the
    // chunks it issued itself, so s_wait_asynccnt <= (7-j) covers chunk j.
    const float4* sv = reinterpret_cast<const float4*>(smem) + tid;
    float ss = 0.0f;
#define SQ_CHUNK(J, N)                                              \
    asm volatile("s_wait_asynccnt " #N ::: "memory");               \
    {                                                               \
        float4 x = sv[(J) * 1024];                                  \
        ss = fmaf(x.x, x.x, ss); ss = fmaf(x.y, x.y, ss);           \
        ss = fmaf(x.z, x.z, ss); ss = fmaf(x.w, x.w, ss);           \
    }
    SQ_CHUNK(0, 7) SQ_CHUNK(1, 6) SQ_CHUNK(2, 5) SQ_CHUNK(3, 4)
    SQ_CHUNK(4, 3) SQ_CHUNK(5, 2) SQ_CHUNK(6, 1) SQ_CHUNK(7, 0)
#undef SQ_CHUNK

    // block_reduce barriers also make every wave's DMA'd data LDS-visible
    const float sumsq = block_reduce_sum(ss, scratch);
    const float norm  = fmaxf(sqrtf(sumsq), 1e-12f);
    const float s     = SCALE_F / norm;

    const int   t  = tgt[row];
    const float xt = smem[t];

    // ---- pass 2: sum of exp(s*x). |logit| <= 30 so no max-shift needed ----
    float se = 0.0f;
#pragma unroll
    for (int j = 0; j < 8; ++j) {
        float4 x = sv[j * 1024];
        se += expf(s * x.x) + expf(s * x.y) + expf(s * x.z) + expf(s * x.w);
    }
    float sumexp = block_reduce_sum(se, scratch);

    if (tid == 0) row_tail(sumexp, xt, s, norm, rowloss, row);
}

// ---------------- generic fallback (any C multiple of 4) ----------------
__global__ void __launch_bounds__(1024)
arcface_focal_row_generic(const float* __restrict__ in,
                          const int*  __restrict__ tgt,
                          float* __restrict__ rowloss,
                          int C)
{
    extern __shared__ float smem[];
    float* scratch = smem + C;

    const int row      = blockIdx.x;
    const int tid      = threadIdx.x;
    const int nthreads = (int)blockDim.x;
    const int nvec     = C >> 2;

    const float* rowp = in + (size_t)row * (size_t)C;

    for (int v = tid; v < nvec; v += nthreads) {
        unsigned lds_addr = (unsigned)(unsigned long long)(const void*)(smem + 4 * v);
        unsigned long long gaddr = (unsigned long long)(const void*)(rowp + 4 * v);
        asm volatile("global_load_async_to_lds_b128 %0, %1, off"
                     :: "v"(lds_addr), "v"(gaddr) : "memory");
    }
    asm volatile("s_wait_asynccnt 0x0" ::: "memory");
    __syncthreads();

    float ss = 0.0f;
    for (int v = tid; v < nvec; v += nthreads) {
        float4 x = reinterpret_cast<const float4*>(smem)[v];
        ss = fmaf(x.x, x.x, ss); ss = fmaf(x.y, x.y, ss);
        ss = fmaf(x.z, x.z, ss); ss = fmaf(x.w, x.w, ss);
    }
    const float sumsq = block_reduce_sum(ss, scratch);
    const float norm  = fmaxf(sqrtf(sumsq), 1e-12f);
    const float s     = SCALE_F / norm;

    const int   t  = tgt[row];
    const float xt = smem[t];

    float se = 0.0f;
    for (int v = tid; v < nvec; v += nthreads) {
        float4 x = reinterpret_cast<const float4*>(smem)[v];
        se += expf(s * x.x) + expf(s * x.y) + expf(s * x.z) + expf(s * x.w);
    }
    float sumexp = block_reduce_sum(se, scratch);

    if (tid == 0) row_tail(sumexp, xt, s, norm, rowloss, row);
}

__global__ void __launch_bounds__(1024)
final_mean_kernel(const float* __restrict__ rowloss, float* __restrict__ out, int B)
{
    __shared__ float scratch[64];
    const int tid = threadIdx.x;
    float v = 0.0f;
    for (int i = tid; i < B; i += (int)blockDim.x) v += rowloss[i];
    const float total = block_reduce_sum(v, scratch);
    if (tid == 0) out[0] = total / (float)B;
}

extern "C" void kernel_launch(void* const* d_in, const int* in_sizes, int n_in,
                              void* d_out, int out_size, void* d_ws, size_t ws_size,
                              hipStream_t stream) {
    (void)n_in; (void)out_size; (void)ws_size;
    const float* in  = (const float*)d_in[0];
    const int*   tgt = (const int*)d_in[1];
    float*       out = (float*)d_out;

    const int B = in_sizes[1];               // 4096 rows
    const int C = in_sizes[0] / B;           // 32768 classes

    float* rowloss = (float*)d_ws;           // B floats of scratch

    const int    threads = 1024;
    const size_t shmem   = (size_t)C * sizeof(float) + 64 * sizeof(float); // 129 KB

    if (C == 32768) {
        hipFuncSetAttribute((const void*)arcface_focal_row_32k,
                            hipFuncAttributeMaxDynamicSharedMemorySize, (int)shmem);
        arcface_focal_row_32k<<<B, threads, shmem, stream>>>(in, tgt, rowloss);
    } else {
        hipFuncSetAttribute((const void*)arcface_focal_row_generic,
                            hipFuncAttributeMaxDynamicSharedMemorySize, (int)shmem);
        arcface_focal_row_generic<<<B, threads, shmem, stream>>>(in, tgt, rowloss, C);
    }
    final_mean_kernel<<<1, threads, 0, stream>>>(rowloss, out, B);
}